// MACE_openmm2_26104811225338
// MI455X (gfx1250) — compile-verified
//
#include <hip/hip_runtime.h>
#include <math.h>

// ---------------- problem constants (from reference) ----------------
#define NA    4096          // atoms
#define F     128           // feature channels
#define KB    8             // radial basis size
#define KF    1024          // K*F
#define MAXN  128           // neighbor cap (mean ~52, 5-sigma ~88)
#define RMAXF 5.0f
#define GAMMA 2.56f         // (K/RMAX)^2 = (8/5)^2
#define MUSTEP 0.714285714285714f // RMAX/(K-1) = 5/7
#define PI_F  3.14159265358979323846f
#define NF    (NA*F)        // 524288

typedef float v2f __attribute__((ext_vector_type(2)));
typedef float v8f __attribute__((ext_vector_type(8)));

__device__ __forceinline__ float sigm(float x){ return 1.0f/(1.0f + expf(-x)); }

// ---------------- h0 = embed[z] ----------------
__global__ void k_embed(const int* __restrict__ z, const float* __restrict__ embed,
                        float* __restrict__ h0){
    int t = blockIdx.x*256 + threadIdx.x;
    int i = t >> 7, f = t & 127;
    h0[t] = embed[z[i]*F + f];
}

// ---------------- neighbor list + rbf (deterministic ballot compaction) ----------------
__global__ void k_build(const float* __restrict__ pos, int* __restrict__ nbr_idx,
                        float* __restrict__ nbr_d, int* __restrict__ nbr_cnt,
                        float* __restrict__ rbf){
    __shared__ float sp[NA*3];                      // 48 KB of the 320 KB LDS
    for (int t = threadIdx.x; t < NA*3; t += 256) sp[t] = pos[t];
    __syncthreads();
    int wv = threadIdx.x >> 5, lane = threadIdx.x & 31;
    int a  = blockIdx.x*8 + wv;
    float ax = sp[a*3+0], ay = sp[a*3+1], az = sp[a*3+2];
    int cnt = 0;
    for (int base = 0; base < NA; base += 32){
        int j = base + lane;
        float dx = ax - sp[j*3+0], dy = ay - sp[j*3+1], dz = az - sp[j*3+2];
        float d2 = dx*dx + dy*dy + dz*dz;
        bool keep = (j != a) && (d2 > 1e-12f) && (d2 < RMAXF*RMAXF);
        unsigned long long bal = __ballot(keep);
        int before = __popcll(bal & ((1ull << lane) - 1ull));
        if (keep){
            int slot = cnt + before;
            if (slot < MAXN){
                float d = sqrtf(d2);
                nbr_idx[a*MAXN + slot] = j;
                nbr_d  [a*MAXN + slot] = d;
                float env = 0.5f*(cosf(PI_F*d/RMAXF) + 1.0f);
                #pragma unroll
                for (int k = 0; k < KB; ++k){
                    float dd = d - (float)k*MUSTEP;
                    rbf[(a*MAXN + slot)*KB + k] = expf(-GAMMA*dd*dd)*env;
                }
            }
        }
        cnt += __popcll(bal);
    }
    if (cnt > MAXN) cnt = MAXN;
    if (lane == 0) nbr_cnt[a] = cnt;
}

// ---------------- forward aggregation: A[i,k,f] = sum_j rbf[i,j,k] h[j,f] ----------------
__global__ void k_agg(const float* __restrict__ rbf, const int* __restrict__ nbr_idx,
                      const int* __restrict__ nbr_cnt, const float* __restrict__ hin,
                      float* __restrict__ Aout){
    int i = blockIdx.x;
    __shared__ float srb[MAXN*KB];
    __shared__ int   sj[MAXN];
    int cnt = nbr_cnt[i];
    for (int t = threadIdx.x; t < cnt*KB; t += 256) srb[t] = rbf[i*MAXN*KB + t];
    for (int t = threadIdx.x; t < cnt;    t += 256) sj[t]  = nbr_idx[i*MAXN + t];
    __syncthreads();
    int f = threadIdx.x & 127, kh = threadIdx.x >> 7;   // kh picks k in [4kh, 4kh+3]
    float a0=0.f, a1=0.f, a2=0.f, a3=0.f;
    for (int s = 0; s < cnt; ++s){
        float hv = hin[sj[s]*F + f];
        const float* r = &srb[s*KB + kh*4];
        a0 += r[0]*hv; a1 += r[1]*hv; a2 += r[2]*hv; a3 += r[3]*hv;
    }
    int b = i*KF + kh*4*F + f;
    Aout[b] = a0; Aout[b+F] = a1; Aout[b+2*F] = a2; Aout[b+3*F] = a3;
}

// ---------------- generic f32 WMMA GEMM: C[M x Nd] = A[M x Kd] * B ----------------
// MODE 0: outH = silu(acc + addm), outS = silu'(...)   (addm may be null)
// MODE 1: outH = acc
// MODE 2: outH = acc * mulm
// BT: B stored [Nd x Kd] row-major (access B[n*ldb + k]); else [Kd x Nd].
template<int MODE, bool BT>
__global__ void k_gemm(const float* __restrict__ A, const float* __restrict__ B,
                       const float* __restrict__ addm, const float* __restrict__ mulm,
                       float* __restrict__ outH, float* __restrict__ outS,
                       int Kd, int Nd, int lda, int ldb){
    int lane = threadIdx.x & 31;
    int wid  = blockIdx.x*8 + (threadIdx.x >> 5);
    int ntn  = Nd >> 4;
    int tm = wid / ntn, tn = wid % ntn;
    int m  = lane & 15, kh = lane >> 4;                 // A layout: lane=16*kh+m
    const float* arow = A + (size_t)(tm*16 + m)*lda;
    v8f c = {0.f,0.f,0.f,0.f,0.f,0.f,0.f,0.f};
    for (int kk = 0; kk < Kd; kk += 4){
        int k0 = kk + 2*kh;
        v2f a; a.x = arow[k0]; a.y = arow[k0+1];
        v2f b;
        if (BT){
            const float* brow = B + (size_t)(tn*16 + m)*ldb;
            b.x = brow[k0]; b.y = brow[k0+1];
        } else {
            b.x = B[(size_t)k0*ldb + tn*16 + m];
            b.y = B[(size_t)(k0+1)*ldb + tn*16 + m];
        }
        c = __builtin_amdgcn_wmma_f32_16x16x4_f32(false, a, false, b, (short)0, c, false, false);
    }
    #pragma unroll
    for (int v = 0; v < 8; ++v){
        int row = tm*16 + v + 8*kh;                    // C/D layout: row = v + 8*(lane>=16)
        int col = tn*16 + m;
        size_t idx = (size_t)row*Nd + col;
        float x = c[v];
        if (MODE == 0){
            if (addm) x += addm[idx];
            float s = sigm(x);
            outH[idx] = x*s;
            outS[idx] = s*(1.0f + x*(1.0f - s));
        } else if (MODE == 1){
            outH[idx] = x;
        } else {
            outH[idx] = x * mulm[idx];
        }
    }
}

// ---------------- transpose aggregation fused with layer-1 gate ----------------
// gpre1[j,f] = ( sum_{i in nbr(j)} sum_k rbf[j,i,k]*GA2[i,k*F+f]  +  gpre2[j,f] ) * spre1[j,f]
__global__ void k_aggback(const float* __restrict__ rbf, const int* __restrict__ nbr_idx,
                          const int* __restrict__ nbr_cnt, const float* __restrict__ GA2,
                          const float* __restrict__ gpre2, const float* __restrict__ spre1,
                          float* __restrict__ gpre1){
    int i = blockIdx.x;
    __shared__ float srb[MAXN*KB];
    __shared__ int   sj[MAXN];
    int cnt = nbr_cnt[i];
    for (int t = threadIdx.x; t < cnt*KB; t += 128) srb[t] = rbf[i*MAXN*KB + t];
    for (int t = threadIdx.x; t < cnt;    t += 128) sj[t]  = nbr_idx[i*MAXN + t];
    __syncthreads();
    int f = threadIdx.x;
    float acc = 0.f;
    for (int s = 0; s < cnt; ++s){
        const float* g = &GA2[(size_t)sj[s]*KF + f];
        const float* r = &srb[s*KB];
        #pragma unroll
        for (int k = 0; k < KB; ++k) acc += r[k]*g[k*F];
    }
    size_t idx = (size_t)i*F + f;
    gpre1[idx] = (acc + gpre2[idx]) * spre1[idx];
}

// ---------------- v[i,g] = su[i,g] * Wr2[g] ----------------
__global__ void k_vmul(const float* __restrict__ su, const float* __restrict__ Wr2,
                       float* __restrict__ v){
    int t = blockIdx.x*256 + threadIdx.x;
    v[t] = su[t] * Wr2[t & 127];
}

// ---------------- energy = sum_{i,g} u[i,g]*Wr2[g]  (single block, deterministic) ----------------
__global__ void k_energy(const float* __restrict__ u, const float* __restrict__ Wr2,
                         float* __restrict__ out){
    __shared__ float red[256];
    float acc = 0.f;
    for (int t = threadIdx.x; t < NF; t += 256) acc += u[t]*Wr2[t & 127];
    red[threadIdx.x] = acc; __syncthreads();
    for (int s = 128; s > 0; s >>= 1){
        if (threadIdx.x < s) red[threadIdx.x] += red[threadIdx.x + s];
        __syncthreads();
    }
    if (threadIdx.x == 0) out[0] = red[0];
}

// ---------------- forces ----------------
// F_a = -sum_{j in nbr(a)} (gd(a,j)+gd(j,a)) * (p_a - p_j)/d
// gd(a,j) = sum_{k,f} drbfdd_k * ( GA2[a,k,f]h1[j,f] + GA1[a,k,f]h0[j,f] )
__global__ void k_force(const float* __restrict__ pos, const int* __restrict__ nbr_idx,
                        const float* __restrict__ nbr_d, const int* __restrict__ nbr_cnt,
                        const float* __restrict__ GA1, const float* __restrict__ GA2,
                        const float* __restrict__ h0, const float* __restrict__ h1,
                        float* __restrict__ out){
    int a = blockIdx.x;
    int lane = threadIdx.x & 31, wv = threadIdx.x >> 5;
    __shared__ float sfx[8], sfy[8], sfz[8];
    int cnt = nbr_cnt[a];
    float ax = pos[a*3+0], ay = pos[a*3+1], az = pos[a*3+2];
    float fx = 0.f, fy = 0.f, fz = 0.f;
    const float* ga2a = GA2 + (size_t)a*KF;
    const float* ga1a = GA1 + (size_t)a*KF;
    const float* h1a  = h1  + (size_t)a*F;
    const float* h0a  = h0  + (size_t)a*F;
    for (int s = wv; s < cnt; s += 8){
        int   j = nbr_idx[a*MAXN + s];
        float d = nbr_d [a*MAXN + s];
        float env  =  0.5f*(cosf(PI_F*d/RMAXF) + 1.0f);
        float denv = -0.5f*(PI_F/RMAXF)*sinf(PI_F*d/RMAXF);
        float drb[KB];
        #pragma unroll
        for (int k = 0; k < KB; ++k){
            float dd = d - (float)k*MUSTEP;
            float e  = expf(-GAMMA*dd*dd);
            drb[k] = e*(-2.0f*GAMMA*dd)*env + e*denv;
        }
        const float* ga2j = GA2 + (size_t)j*KF;
        const float* ga1j = GA1 + (size_t)j*KF;
        const float* h1j  = h1  + (size_t)j*F;
        const float* h0j  = h0  + (size_t)j*F;
        float acc = 0.f;
        #pragma unroll
        for (int it = 0; it < 32; ++it){          // k = it>>2 is compile-time constant
            const int k = it >> 2;
            int f = (it & 3)*32 + lane;
            int idx = k*F + f;
            float w = drb[k];
            acc += w*( ga2a[idx]*h1j[f] + ga1a[idx]*h0j[f]
                     + ga2j[idx]*h1a[f] + ga1j[idx]*h0a[f] );
        }
        #pragma unroll
        for (int off = 16; off > 0; off >>= 1) acc += __shfl_xor(acc, off, 32);
        if (lane == 0){
            float coef = acc / d;
            fx += coef*(ax - pos[j*3+0]);
            fy += coef*(ay - pos[j*3+1]);
            fz += coef*(az - pos[j*3+2]);
        }
    }
    if (lane == 0){ sfx[wv] = fx; sfy[wv] = fy; sfz[wv] = fz; }
    __syncthreads();
    if (threadIdx.x == 0){
        float tx=0.f, ty=0.f, tz=0.f;
        #pragma unroll
        for (int w = 0; w < 8; ++w){ tx += sfx[w]; ty += sfy[w]; tz += sfz[w]; }
        out[1 + a*3 + 0] = -tx;
        out[1 + a*3 + 1] = -ty;
        out[1 + a*3 + 2] = -tz;
    }
}

// ---------------- host-side launch ----------------
extern "C" void kernel_launch(void* const* d_in, const int* in_sizes, int n_in,
                              void* d_out, int out_size, void* d_ws, size_t ws_size,
                              hipStream_t stream){
    const float* pos   = (const float*)d_in[0];
    const int*   z     = (const int*)  d_in[1];
    const float* embed = (const float*)d_in[2];
    const float* W1    = (const float*)d_in[3];   // [K,F,F] flat = [KF x F]
    const float* W2    = (const float*)d_in[4];
    const float* Wr1   = (const float*)d_in[5];   // [F x F]
    const float* Wr2   = (const float*)d_in[6];   // [F]
    float* out = (float*)d_out;                   // [0]=energy, [1..]=forces

    // workspace carve-out (~92 MB total)
    char* w = (char*)d_ws;
    auto carve = [&](size_t nbytes)->void*{
        void* p = (void*)w; w += ((nbytes + 255) & ~(size_t)255); return p;
    };
    float* h0    = (float*)carve((size_t)NF*4);
    float* h1    = (float*)carve((size_t)NF*4);
    float* h2    = (float*)carve((size_t)NF*4);
    float* u     = (float*)carve((size_t)NF*4);
    float* su    = (float*)carve((size_t)NF*4);
    float* spre1 = (float*)carve((size_t)NF*4);
    float* spre2 = (float*)carve((size_t)NF*4);
    float* gpre2 = (float*)carve((size_t)NF*4);
    float* gpre1 = (float*)carve((size_t)NF*4);
    float* vbuf  = (float*)carve((size_t)NF*4);
    float* Abuf  = (float*)carve((size_t)NA*KF*4);
    float* GA2   = (float*)carve((size_t)NA*KF*4);
    float* GA1   = (float*)carve((size_t)NA*KF*4);
    float* rbf   = (float*)carve((size_t)NA*MAXN*KB*4);
    float* nbrd  = (float*)carve((size_t)NA*MAXN*4);
    int*   nbri  = (int*)  carve((size_t)NA*MAXN*4);
    int*   nbrc  = (int*)  carve((size_t)NA*4);

    // ---- forward ----
    k_embed<<<NF/256, 256, 0, stream>>>(z, embed, h0);
    k_build<<<NA/8, 256, 0, stream>>>(pos, nbri, nbrd, nbrc, rbf);

    k_agg<<<NA, 256, 0, stream>>>(rbf, nbri, nbrc, h0, Abuf);
    k_gemm<0,false><<<(NA/16)*(F/16)/8, 256, 0, stream>>>(Abuf, W1, h0, nullptr,
                                                          h1, spre1, KF, F, KF, F);
    k_agg<<<NA, 256, 0, stream>>>(rbf, nbri, nbrc, h1, Abuf);
    k_gemm<0,false><<<(NA/16)*(F/16)/8, 256, 0, stream>>>(Abuf, W2, h1, nullptr,
                                                          h2, spre2, KF, F, KF, F);
    // readout: t = h2 @ Wr1 ; u = silu(t), su = silu'(t)
    k_gemm<0,false><<<(NA/16)*(F/16)/8, 256, 0, stream>>>(h2, Wr1, nullptr, nullptr,
                                                          u, su, F, F, F, F);
    k_energy<<<1, 256, 0, stream>>>(u, Wr2, out);

    // ---- backward ----
    k_vmul<<<NF/256, 256, 0, stream>>>(su, Wr2, vbuf);
    // gpre2 = (vbuf @ Wr1^T) * spre2
    k_gemm<2,true><<<(NA/16)*(F/16)/8, 256, 0, stream>>>(vbuf, Wr1, nullptr, spre2,
                                                         gpre2, nullptr, F, F, F, F);
    // GA2 = gpre2 @ W2^T  -> [NA x KF]
    k_gemm<1,true><<<(NA/16)*(KF/16)/8, 256, 0, stream>>>(gpre2, W2, nullptr, nullptr,
                                                          GA2, nullptr, F, KF, F, F);
    // gpre1 = (rbf^T-aggregate(GA2) + gpre2) * spre1
    k_aggback<<<NA, 128, 0, stream>>>(rbf, nbri, nbrc, GA2, gpre2, spre1, gpre1);
    // GA1 = gpre1 @ W1^T
    k_gemm<1,true><<<(NA/16)*(KF/16)/8, 256, 0, stream>>>(gpre1, W1, nullptr, nullptr,
                                                          GA1, nullptr, F, KF, F, F);
    // forces
    k_force<<<NA, 256, 0, stream>>>(pos, nbri, nbrd, nbrc, GA1, GA2, h0, h1, out);
}